// FedProxLoss_39608188404052
// MI455X (gfx1250) — compile-verified
//
#include <hip/hip_runtime.h>
#include <hip/hip_bf16.h>
#include <math.h>

// ---------------------------------------------------------------------------
// FedProx loss for MI455X (gfx1250, wave32).
// Bandwidth-bound: ~671MB streamed once -> b128 non-temporal loads.
// Sum-of-squares accumulated EXACTLY in f32 on the matrix pipe via
// V_WMMA_F32_16X16X4_F32 with an all-ones B matrix (D = A*1 + C).
// Deterministic two-stage reduction (block partials in d_ws, fixed-order
// final reduce), no float atomics.
// ---------------------------------------------------------------------------

typedef float f4  __attribute__((ext_vector_type(4)));
typedef float v2f __attribute__((ext_vector_type(2)));
typedef float v8f __attribute__((ext_vector_type(8)));

#define PROX_HAS_F32_WMMA __has_builtin(__builtin_amdgcn_wmma_f32_16x16x4_f32)

// d_ws float layout: 4 regions of 2048 partials (one per param pair),
// then 64 BCE partials.
#define PAIR_REGION   2048
#define BCE_BASE      (4 * PAIR_REGION)
#define BCE_BLOCKS    64

// ---------------------------------------------------------------------------
// Streaming sum((lp-gp)^2) over one tensor pair. n4 = element count / 4.
// n4 is a multiple of 32 for every tensor here, so the grid-stride loop
// condition is wave-uniform -> EXEC stays all-ones around the WMMA.
// ---------------------------------------------------------------------------
__global__ __launch_bounds__(256) void prox_sumsq_kernel(
    const float* __restrict__ lp, const float* __restrict__ gp,
    unsigned int n4, float* __restrict__ partials) {
  const unsigned int tid    = blockIdx.x * blockDim.x + threadIdx.x;
  const unsigned int stride = gridDim.x * blockDim.x;
  const f4* __restrict__ lp4 = (const f4*)lp;
  const f4* __restrict__ gp4 = (const f4*)gp;

#if PROX_HAS_F32_WMMA
  v8f c = {0.f, 0.f, 0.f, 0.f, 0.f, 0.f, 0.f, 0.f};
  const v2f ones = {1.0f, 1.0f};   // B matrix = all ones (4x16)
  for (unsigned int i = tid; i < n4; i += stride) {
    f4 l = __builtin_nontemporal_load(lp4 + i);   // global_load_b128, TH=NT
    f4 g = __builtin_nontemporal_load(gp4 + i);
    f4 d = l - g;
    v2f a0 = { d.x * d.x, d.y * d.y };
    v2f a1 = { d.z * d.z, d.w * d.w };
    // D = A x Ones + C : accumulates sum of all A entries into C rows (f32 exact)
    c = __builtin_amdgcn_wmma_f32_16x16x4_f32(false, a0, false, ones,
                                              (short)0, c, false, false);
    c = __builtin_amdgcn_wmma_f32_16x16x4_f32(false, a1, false, ones,
                                              (short)0, c, false, false);
  }
  // Every column of D is identical (B = ones). Lanes 0..15 hold row sums
  // M=0..7 in c[0..7]; lanes 16..31 hold M=8..15. Wave total = lane0 + lane16.
  float s = c[0] + c[1] + c[2] + c[3] + c[4] + c[5] + c[6] + c[7];
  float waveSum = __shfl(s, 0, 32) + __shfl(s, 16, 32);
#else
  float acc = 0.f;
  for (unsigned int i = tid; i < n4; i += stride) {
    f4 l = __builtin_nontemporal_load(lp4 + i);
    f4 g = __builtin_nontemporal_load(gp4 + i);
    f4 d = l - g;
    acc = fmaf(d.x, d.x, acc);
    acc = fmaf(d.y, d.y, acc);
    acc = fmaf(d.z, d.z, acc);
    acc = fmaf(d.w, d.w, acc);
  }
  float waveSum = acc;
  for (int off = 16; off > 0; off >>= 1)
    waveSum += __shfl_xor(waveSum, off, 32);
#endif

  // Block reduce (8 waves of 32).
  __shared__ float lds[8];
  const int lane = threadIdx.x & 31;
  const int wave = threadIdx.x >> 5;
  if (lane == 0) lds[wave] = waveSum;
  __syncthreads();
  if (threadIdx.x == 0) {
    float b = 0.f;
    #pragma unroll
    for (int w = 0; w < 8; ++w) b += lds[w];
    partials[blockIdx.x] = b;   // fixed slot -> deterministic
  }
}

// ---------------------------------------------------------------------------
// Sum of the three BCE-with-logits means (weights all 1.0).
// 3 * 16384 elements; each block writes one partial (already /16384 scaled).
// ---------------------------------------------------------------------------
__global__ __launch_bounds__(256) void bce_kernel(
    const float* __restrict__ p0, const float* __restrict__ p1,
    const float* __restrict__ p2, const float* __restrict__ t0,
    const float* __restrict__ t1, const float* __restrict__ t2,
    float* __restrict__ partials) {
  const unsigned int N   = 16384u;
  const unsigned int tot = 3u * N;
  const unsigned int tid    = blockIdx.x * blockDim.x + threadIdx.x;
  const unsigned int stride = gridDim.x * blockDim.x;
  float acc = 0.f;
  for (unsigned int i = tid; i < tot; i += stride) {
    const unsigned int task = i >> 14;        // i / 16384
    const unsigned int j    = i & (N - 1u);   // i % 16384
    const float* pp = (task == 0u) ? p0 : (task == 1u) ? p1 : p2;
    const float* tt = (task == 0u) ? t0 : (task == 1u) ? t1 : t2;
    const float x = pp[j];
    const float y = tt[j];
    acc += fmaxf(x, 0.f) - x * y + log1pf(__expf(-fabsf(x)));
  }
  acc *= (1.0f / 16384.0f);
  for (int off = 16; off > 0; off >>= 1)
    acc += __shfl_xor(acc, off, 32);
  __shared__ float lds[8];
  const int lane = threadIdx.x & 31;
  const int wave = threadIdx.x >> 5;
  if (lane == 0) lds[wave] = acc;
  __syncthreads();
  if (threadIdx.x == 0) {
    float b = 0.f;
    #pragma unroll
    for (int w = 0; w < 8; ++w) b += lds[w];
    partials[blockIdx.x] = b;
  }
}

// ---------------------------------------------------------------------------
// Final deterministic combine: total = (MU/2)*sum(prox partials) + sum(bce).
// Only sums the slots the earlier launches actually wrote.
// ---------------------------------------------------------------------------
__global__ __launch_bounds__(256) void finalize_kernel(
    const float* __restrict__ ws, float* __restrict__ out,
    int b0, int b1, int b2, int b3) {
  const int cnt[4] = {b0, b1, b2, b3};
  float sp = 0.f;
  for (int r = 0; r < 4; ++r) {
    const float* base = ws + r * PAIR_REGION;
    for (int i = (int)threadIdx.x; i < cnt[r]; i += 256) sp += base[i];
  }
  float sb = 0.f;
  for (int i = (int)threadIdx.x; i < BCE_BLOCKS; i += 256) sb += ws[BCE_BASE + i];
  float s = 0.005f * sp + sb;   // MU/2 = 0.005

  __shared__ float lds[256];
  lds[threadIdx.x] = s;
  __syncthreads();
  for (int off = 128; off > 0; off >>= 1) {
    if ((int)threadIdx.x < off) lds[threadIdx.x] += lds[threadIdx.x + off];
    __syncthreads();
  }
  if (threadIdx.x == 0) out[0] = lds[0];
}

// ---------------------------------------------------------------------------
extern "C" void kernel_launch(void* const* d_in, const int* in_sizes, int n_in,
                              void* d_out, int out_size, void* d_ws, size_t ws_size,
                              hipStream_t stream) {
  (void)n_in; (void)out_size; (void)ws_size;
  // setup_inputs() order: preds0..2, targets0..2, local_p0..3, global_p0..3
  const float* preds[3]   = {(const float*)d_in[0], (const float*)d_in[1], (const float*)d_in[2]};
  const float* targets[3] = {(const float*)d_in[3], (const float*)d_in[4], (const float*)d_in[5]};
  const float* localp[4]  = {(const float*)d_in[6], (const float*)d_in[7],
                             (const float*)d_in[8], (const float*)d_in[9]};
  const float* globalp[4] = {(const float*)d_in[10], (const float*)d_in[11],
                             (const float*)d_in[12], (const float*)d_in[13]};
  float* ws  = (float*)d_ws;
  float* out = (float*)d_out;

  int blocks[4];
  for (int r = 0; r < 4; ++r) {
    const unsigned int n4 = (unsigned int)(in_sizes[6 + r] / 4);
    unsigned int b = (n4 + 255u) / 256u;
    if (b > (unsigned)PAIR_REGION) b = PAIR_REGION;
    blocks[r] = (int)b;
    prox_sumsq_kernel<<<b, 256, 0, stream>>>(localp[r], globalp[r], n4,
                                             ws + r * PAIR_REGION);
  }

  bce_kernel<<<BCE_BLOCKS, 256, 0, stream>>>(preds[0], preds[1], preds[2],
                                             targets[0], targets[1], targets[2],
                                             ws + BCE_BASE);

  finalize_kernel<<<1, 256, 0, stream>>>(ws, out, blocks[0], blocks[1],
                                         blocks[2], blocks[3]);
}